// SwitchableNorm2d_Unholy_16295105921565
// MI455X (gfx1250) — compile-verified
//
#include <hip/hip_runtime.h>
#include <math.h>

// ---------------------------------------------------------------------------
// SwitchableNorm2d (training mode) for x[N=32, C=256, H=64, W=64] fp32.
// Memory-bound: 2 reads + 1 write of 128 MiB  =>  ~11-17 us floor @ 23.3 TB/s.
// Pass 1 computes per-(n,c) sum / sumsq with a WMMA ones-matrix reduction.
// Pass 2 folds BN/IN/LN stats + double-softmax mixing into per-plane scale/shift.
// Pass 3 is a single FMA per element with nontemporal stores.
// ---------------------------------------------------------------------------

typedef __attribute__((ext_vector_type(2))) float v2f;
typedef __attribute__((ext_vector_type(4))) float v4f;
typedef __attribute__((ext_vector_type(8))) float v8f;

#define SN_N   32
#define SN_C   256
#define SN_HW  4096            // 64*64
#define SN_P   (SN_N * SN_C)   // 8192 planes
#define SN_EPS 1e-5f

// ---------------------------------------------------------------------------
// Pass 1: per-(n,c) plane sum and sum-of-squares.
// One 256-thread block (8 waves) per plane; each thread: 4x b128 loads.
// Wave reduce: A = lane partials (vgpr0=val, vgpr1=0), B = ones(4x16),
// D = A x B:  rowsum_m = s_m + s_{m+16}, replicated across D columns.
// Per lane: sum of 8 D VGPRs = half-sum; shfl_xor(16) completes 32-lane sum.
// ---------------------------------------------------------------------------
__global__ __launch_bounds__(256) void sn2d_stats_kernel(
    const float* __restrict__ x,
    float* __restrict__ in_sum,
    float* __restrict__ in_sumsq) {
  const int p = blockIdx.x;   // plane index = n*C + c
  const int t = threadIdx.x;  // 0..255

  const v4f* __restrict__ xv = (const v4f*)(x + (size_t)p * SN_HW);

  float s = 0.0f, q = 0.0f;
#pragma unroll
  for (int i = 0; i < 4; ++i) {
    v4f v = xv[i * 256 + t];
    s += v.x + v.y + v.z + v.w;
    q = __builtin_fmaf(v.x, v.x, q);
    q = __builtin_fmaf(v.y, v.y, q);
    q = __builtin_fmaf(v.z, v.z, q);
    q = __builtin_fmaf(v.w, v.w, q);
  }

  // --- WMMA 32-lane reduction (EXEC is all ones here: no divergence yet) ---
  v2f aS; aS.x = s; aS.y = 0.0f;
  v2f aQ; aQ.x = q; aQ.y = 0.0f;
  v2f ones; ones.x = 1.0f; ones.y = 1.0f;
  v8f zero = {};

  v8f dS = __builtin_amdgcn_wmma_f32_16x16x4_f32(
      false, aS, false, ones, (short)0, zero, false, false);
  v8f dQ = __builtin_amdgcn_wmma_f32_16x16x4_f32(
      false, aQ, false, ones, (short)0, zero, false, false);

  float hs = dS.s0 + dS.s1 + dS.s2 + dS.s3 + dS.s4 + dS.s5 + dS.s6 + dS.s7;
  float hq = dQ.s0 + dQ.s1 + dQ.s2 + dQ.s3 + dQ.s4 + dQ.s5 + dQ.s6 + dQ.s7;

  hs += __shfl_xor(hs, 16);   // combine row-blocks 0..7 and 8..15
  hq += __shfl_xor(hq, 16);

  __shared__ float ls[8], lq[8];
  const int wave = t >> 5;
  const int lane = t & 31;
  if (lane == 0) { ls[wave] = hs; lq[wave] = hq; }
  __syncthreads();

  if (t == 0) {
    float S = 0.0f, Q = 0.0f;
#pragma unroll
    for (int w = 0; w < 8; ++w) { S += ls[w]; Q += lq[w]; }
    in_sum[p]   = S;
    in_sumsq[p] = Q;
  }
}

// ---------------------------------------------------------------------------
// Double softmax over 3 elements (applied twice along dim 0).
// ---------------------------------------------------------------------------
__device__ inline void double_softmax3(float a, float b, float c,
                                       float& o0, float& o1, float& o2) {
#pragma unroll
  for (int r = 0; r < 2; ++r) {
    float mx  = fmaxf(a, fmaxf(b, c));
    float ea  = __expf(a - mx), eb = __expf(b - mx), ec = __expf(c - mx);
    float inv = 1.0f / (ea + eb + ec);
    a = ea * inv; b = eb * inv; c = ec * inv;
  }
  o0 = a; o1 = b; o2 = c;
}

// ---------------------------------------------------------------------------
// Pass 2: fold BN/IN/LN stats + mixing weights + affine params into
// per-plane scale/shift.  Single 256-thread block; trivial cost.
// ---------------------------------------------------------------------------
__global__ __launch_bounds__(256) void sn2d_coeff_kernel(
    const float* __restrict__ in_sum,
    const float* __restrict__ in_sumsq,
    const float* __restrict__ weight,
    const float* __restrict__ bias,
    const float* __restrict__ mean_w,
    const float* __restrict__ var_w,
    float* __restrict__ scale_arr,
    float* __restrict__ shift_arr) {
  __shared__ float mbn[SN_C], vbn[SN_C], mln[SN_N], vln[SN_N];
  const int t = threadIdx.x;  // 0..255

  // BN stats for channel c = t  (reduce over n; coalesced across threads)
  {
    float s = 0.0f, q = 0.0f;
#pragma unroll 4
    for (int n = 0; n < SN_N; ++n) {
      s += in_sum[n * SN_C + t];
      q += in_sumsq[n * SN_C + t];
    }
    const float inv = 1.0f / (float)(SN_N * SN_HW);   // 1/131072
    float m = s * inv;
    mbn[t] = m;
    vbn[t] = __builtin_fmaf(-m, m, q * inv);
  }

  // LN stats for sample n = t (first 32 threads)
  if (t < SN_N) {
    float s = 0.0f, q = 0.0f;
    for (int c = 0; c < SN_C; ++c) {
      int p = t * SN_C + c;
      s += in_sum[p];
      q += in_sumsq[p];
    }
    const float inv = 1.0f / (float)(SN_C * SN_HW);   // 1/1048576
    float m = s * inv;
    mln[t] = m;
    vln[t] = __builtin_fmaf(-m, m, q * inv);
  }
  __syncthreads();

  float mw0, mw1, mw2, vw0, vw1, vw2;
  double_softmax3(mean_w[0], mean_w[1], mean_w[2], mw0, mw1, mw2);
  double_softmax3(var_w[0],  var_w[1],  var_w[2],  vw0, vw1, vw2);

  const float wgt = weight[t];  // channel c == t in the loop below
  const float bia = bias[t];
  const float inv_hw = 1.0f / (float)SN_HW;

  // plane p = n*256 + t  =>  n = k, c = t
  for (int k = 0; k < SN_N; ++k) {
    int p = k * SN_C + t;
    float mi = in_sum[p] * inv_hw;
    float vi = __builtin_fmaf(-mi, mi, in_sumsq[p] * inv_hw);

    float mean = mw0 * mbn[t] + mw1 * mi + mw2 * mln[k];
    float var  = vw0 * vbn[t] + vw1 * vi + vw2 * vln[k] + SN_EPS;

    float sc = wgt / sqrtf(var);
    scale_arr[p] = sc;
    shift_arr[p] = __builtin_fmaf(-mean, sc, bia);
  }
}

// ---------------------------------------------------------------------------
// Pass 3: out = x * scale[p] + shift[p].  One block per plane; b128 loads
// (likely L2 hits since x fits in 192MB L2), nontemporal b128 stores so the
// output stream does not evict x.
// ---------------------------------------------------------------------------
__global__ __launch_bounds__(256) void sn2d_apply_kernel(
    const float* __restrict__ x,
    const float* __restrict__ scale_arr,
    const float* __restrict__ shift_arr,
    float* __restrict__ out) {
  const int p = blockIdx.x;
  const int t = threadIdx.x;

  const float sc = scale_arr[p];   // uniform -> scalarized by compiler
  const float sh = shift_arr[p];

  const v4f* __restrict__ xv = (const v4f*)(x + (size_t)p * SN_HW);
  v4f* __restrict__ ov = (v4f*)(out + (size_t)p * SN_HW);

#pragma unroll
  for (int i = 0; i < 4; ++i) {
    v4f v = xv[i * 256 + t];
    v4f r;
    r.x = __builtin_fmaf(v.x, sc, sh);
    r.y = __builtin_fmaf(v.y, sc, sh);
    r.z = __builtin_fmaf(v.z, sc, sh);
    r.w = __builtin_fmaf(v.w, sc, sh);
    __builtin_nontemporal_store(r, &ov[i * 256 + t]);
  }
}

// ---------------------------------------------------------------------------
// Launch
// ---------------------------------------------------------------------------
extern "C" void kernel_launch(void* const* d_in, const int* in_sizes, int n_in,
                              void* d_out, int out_size, void* d_ws, size_t ws_size,
                              hipStream_t stream) {
  const float* x       = (const float*)d_in[0];
  const float* weight  = (const float*)d_in[1];
  const float* bias    = (const float*)d_in[2];
  const float* mean_w  = (const float*)d_in[3];
  const float* var_w   = (const float*)d_in[4];
  float* out = (float*)d_out;

  float* ws        = (float*)d_ws;
  float* in_sum    = ws;               // 8192
  float* in_sumsq  = ws + SN_P;        // 8192
  float* scale_arr = ws + 2 * SN_P;    // 8192
  float* shift_arr = ws + 3 * SN_P;    // 8192

  sn2d_stats_kernel<<<SN_P, 256, 0, stream>>>(x, in_sum, in_sumsq);
  sn2d_coeff_kernel<<<1, 256, 0, stream>>>(in_sum, in_sumsq, weight, bias,
                                           mean_w, var_w, scale_arr, shift_arr);
  sn2d_apply_kernel<<<SN_P, 256, 0, stream>>>(x, scale_arr, shift_arr, out);
}